// RayMarcherModule_10814727652054
// MI455X (gfx1250) — compile-verified
//
#include <hip/hip_runtime.h>
#include <hip/hip_bf16.h>

// ---------------------------------------------------------------------------
// Volumetric ray marcher, 1024x1024, one thread per pixel (wave32, gfx1250).
// Transcendental-bound: 14x v_exp_f32 + 13x v_sqrt_f32 per pixel, fully
// unrolled so the trans pipe co-executes with VALU. Memory is a single
// streaming b96 load + b96 store per lane (~24 MB total, ~1us at 23.3 TB/s),
// far below the compute time, so no LDS/TDM staging is warranted.
// ---------------------------------------------------------------------------

#define MAX_STEPS   12
#define INV_STEP    5.0f      // 1 / BASE_STEP (0.2)
#define SIGMA       0.6f      // SCATTERING + ABSORPTION + 0.4
#define LIGHT_K     1.2f      // DENSITY * (SCATTERING + ABSORPTION)
#define RES_K       0.6f      // SCATTERING * DENSITY
#define RO_X        0.0f
#define RO_Y        0.0f
#define RO_Z        3.0f
#define BG_R        0.572f
#define BG_G        0.772f
#define BG_B        0.921f
#define LC_R        1.3f
#define LC_G        0.3f
#define LC_B        0.9f

__device__ __forceinline__ float fast_exp(float x) {
    // e^x = 2^(x * log2(e)) -> one v_mul_f32 + v_exp_f32 (hw transcendental)
    return __builtin_amdgcn_exp2f(x * 1.44269504088896340736f);
}
__device__ __forceinline__ float fast_sqrt(float x) {
    return __builtin_amdgcn_sqrtf(x);   // raw v_sqrt_f32 (~1 ulp)
}
__device__ __forceinline__ float fast_rcp(float x) {
    return __builtin_amdgcn_rcpf(x);    // raw v_rcp_f32 (~1 ulp)
}

__global__ __launch_bounds__(256) void raymarch_kernel(
    const float* __restrict__ scene,   // [4]  : cx, cy, cz, radius
    const float* __restrict__ dirs,    // [N,3]: ray directions (normalized)
    float*       __restrict__ out,     // [N,3]: pixels
    int npix)
{
    const int i = blockIdx.x * blockDim.x + threadIdx.x;
    if (i >= npix) return;

    // Uniform scene constant: address is lane-invariant -> s_load_b128.
    const float4 sc = *reinterpret_cast<const float4*>(scene);
    const float cx = sc.x, cy = sc.y, cz = sc.z, radius = sc.w;
    const float r2 = radius * radius;

    // Per-pixel ray direction: 3 adjacent floats -> global_load_b96.
    const float rdx = dirs[3 * i + 0];
    const float rdy = dirs[3 * i + 1];
    const float rdz = dirs[3 * i + 2];

    // oc = RO - center
    const float ocx = RO_X - cx;
    const float ocy = RO_Y - cy;
    const float ocz = RO_Z - cz;

    const float a    = rdx * rdx + rdy * rdy + rdz * rdz;
    const float b    = 2.0f * (rdx * ocx + rdy * ocy + rdz * ocz);
    const float c    = ocx * ocx + ocy * ocy + ocz * ocz - r2;
    const float disc = b * b - 4.0f * a * c;

    // Reference semantics: sqrt argument uses strict disc > 0, else 1.0.
    const float sq    = fast_sqrt(disc > 0.0f ? disc : 1.0f);
    const float inv2a = fast_rcp(2.0f * a);
    const float x1    = (-b - sq) * inv2a;
    const float x2    = (-b + sq) * inv2a;
    const bool  hit   = (disc >= 0.0f) && (x2 >= 0.0f);

    const float t0      = fmaxf(x1, 0.0f);
    const float t1      = x2;
    const float seg     = t1 - t0;
    const float ns      = ceilf(seg * INV_STEP);
    const float ns_safe = fmaxf(ns, 1.0f);
    const float step    = seg * fast_rcp(ns_safe);

    float acc = 0.0f;
#pragma unroll
    for (int k = 0; k < MAX_STEPS; ++k) {
        const float fk    = (float)k;
        const bool  smask = (fk < ns) && hit;

        const float t  = t1 - (fk + 0.5f) * step;
        // p = RO + t*rd ; pc = p - center
        const float pcx = (RO_X + t * rdx) - cx;
        const float pcy = (RO_Y + t * rdy) - cy;
        const float pcz = (RO_Z + t * rdz) - cz;

        const float lb  = 2.0f * pcy;
        const float lc  = pcx * pcx + pcy * pcy + pcz * pcz - r2;
        const float ld  = lb * lb - 4.0f * lc;
        const float lsq = fast_sqrt(ld > 0.0f ? ld : 1.0f);
        const float vt1 = 0.5f * (-lb + lsq);
        const bool  lhit = (ld >= 0.0f) && (vt1 >= 0.0f);

        const float la = fast_exp(-LIGHT_K * vt1);
        acc += (smask && lhit) ? la : 0.0f;
    }

    // s = exp(-SIGMA*step); s**ns = exp(-SIGMA*step*ns); s**(ns+1) = s**ns * s
    const float s            = fast_exp(-SIGMA * step);
    const float transparency = fast_exp(-SIGMA * step * ns);
    const float result       = RES_K * step * transparency * s * acc;

    float pr, pg, pb;
    if (hit) {
        pr = BG_R * transparency + LC_R * result;
        pg = BG_G * transparency + LC_G * result;
        pb = BG_B * transparency + LC_B * result;
    } else {
        pr = BG_R; pg = BG_G; pb = BG_B;
    }

    // 3 adjacent floats -> global_store_b96 (streaming, written once).
    out[3 * i + 0] = pr;
    out[3 * i + 1] = pg;
    out[3 * i + 2] = pb;
}

extern "C" void kernel_launch(void* const* d_in, const int* in_sizes, int n_in,
                              void* d_out, int out_size, void* d_ws, size_t ws_size,
                              hipStream_t stream) {
    (void)n_in; (void)out_size; (void)d_ws; (void)ws_size;

    const float* scene = (const float*)d_in[0];   // [1,4] float32
    const float* dirs  = (const float*)d_in[1];   // [1024,1024,3] float32
    float*       out   = (float*)d_out;           // [1024,1024,3] float32

    const int npix   = in_sizes[1] / 3;
    const int block  = 256;                        // 8 wave32 waves per block
    const int grid   = (npix + block - 1) / block;

    raymarch_kernel<<<grid, block, 0, stream>>>(scene, dirs, out, npix);
}